// vanilla_conv1d_fft_6717328851328
// MI455X (gfx1250) — compile-verified
//
#include <hip/hip_runtime.h>

// Depthwise causal 4-tap conv, stride 4, via V_WMMA_F32_16X16X4_F32.
// x: (4,2048,8192) f32, weight: (2048,1,4) f32, out: (4,2048,2048) f32.
// y[r,o] = sum_k (w[c,k]*0.5) * x[r, 4o-k], zero-padded at row start.

typedef __attribute__((ext_vector_type(2))) float v2f;
typedef __attribute__((ext_vector_type(8))) float v8f;

#define LROW 8192
#define OROW 2048
#define CCH  2048
#define PAD  4

__global__ __launch_bounds__(256) void conv1d_stride4_wmma_kernel(
    const float* __restrict__ x,
    const float* __restrict__ weight,
    float* __restrict__ out)
{
    __shared__ float tile[PAD + LROW];   // tile[PAD + t] == x_row[t]; tile[1..3] = x[-3..-1] = 0

    const int row = blockIdx.x;          // row = b*CCH + c
    const int c   = row & (CCH - 1);
    const float* __restrict__ xrow = x   + (size_t)row * LROW;
    float* __restrict__       orow = out + (size_t)row * OROW;

    // zero the causal halo (tile[0] unused, kept for 16B alignment of the row)
    if (threadIdx.x < PAD) tile[threadIdx.x] = 0.0f;

    // coalesced float4 load of the whole row into LDS (16B-aligned ds_store_b128)
    const float4* __restrict__ x4 = reinterpret_cast<const float4*>(xrow);
    for (int p = threadIdx.x; p < LROW / 4; p += 256) {
        *reinterpret_cast<float4*>(&tile[PAD + p * 4]) = x4[p];
    }
    __syncthreads();

    // per-channel taps, folded scale 1/sqrt(K)=0.5
    const float w0 = weight[c * 4 + 0] * 0.5f;
    const float w1 = weight[c * 4 + 1] * 0.5f;
    const float w2 = weight[c * 4 + 2] * 0.5f;
    const float w3 = weight[c * 4 + 3] * 0.5f;

    const int lane = threadIdx.x & 31;
    const int wv   = threadIdx.x >> 5;   // wave id 0..7
    const int j    = lane & 15;          // N column within the 16-output group
    const int hh   = lane >> 4;          // 0: this lane half carries K={0,1}; 1: K={2,3}

    // A (16x4, SRC0): A[i,k] = w'[k] broadcast over rows i.
    // 32-bit A layout: vgpr0 = K0 (lanes 0-15) / K2 (lanes 16-31); vgpr1 = K1 / K3.
    v2f a;
    a.x = hh ? w2 : w0;
    a.y = hh ? w3 : w1;

    // B (4x16, SRC1): B[k,j] = x[4(G+j) - k].
    // vgpr0 = K0/K2 split by lane half -> tile[(4 or 2) + 4(G+j)]; vgpr1 = K1/K3 -> (3 or 1).
    const int off0 = PAD - 2 * hh;       // K=0 -> x[t]   | K=2 -> x[t-2]
    const int off1 = off0 - 1;           // K=1 -> x[t-1] | K=3 -> x[t-3]

    v8f cz = {};                         // C = 0

    for (int it = 0; it < 8; ++it) {
        const int G     = wv * 256 + it * 32;   // output group base (two 16-groups per iter)
        const int baseA = 4 * (G + j);
        v2f bA, bB;
        bA.x = tile[off0 + baseA];
        bA.y = tile[off1 + baseA];
        bB.x = tile[off0 + baseA + 64];         // group G+16 -> +4*16 input floats
        bB.y = tile[off1 + baseA + 64];

        // D = A x B + 0. All rows of D are identical (A rows identical), so D vgpr0
        // holds y[G + j] in every lane (rows 0 and 8 coincide).
        v8f dA = __builtin_amdgcn_wmma_f32_16x16x4_f32(
            false, a, false, bA, (short)0, cz, false, false);
        v8f dB = __builtin_amdgcn_wmma_f32_16x16x4_f32(
            false, a, false, bB, (short)0, cz, false, false);

        // lanes 0-15 store group G, lanes 16-31 store group G+16: one coalesced b32 store
        orow[G + lane] = hh ? dB[0] : dA[0];
    }
}

extern "C" void kernel_launch(void* const* d_in, const int* in_sizes, int n_in,
                              void* d_out, int out_size, void* d_ws, size_t ws_size,
                              hipStream_t stream) {
    const float* x      = (const float*)d_in[0];   // (4,2048,8192) f32
    const float* weight = (const float*)d_in[1];   // (2048,1,4)   f32
    float*       out    = (float*)d_out;           // (4,2048,2048) f32

    (void)in_sizes; (void)n_in; (void)out_size; (void)d_ws; (void)ws_size;

    const int rows = 4 * CCH;                      // 8192 (b,c) rows
    conv1d_stride4_wmma_kernel<<<rows, 256, 0, stream>>>(x, weight, out);
}